// GATTransformer_89240830476323
// MI455X (gfx1250) — compile-verified
//
#include <hip/hip_runtime.h>
#include <math.h>
#include <stdint.h>

// ---------------------------------------------------------------------------
// GAT(4096 nodes, 131072 edges) + Transformer(d=16) on gfx1250.
// Dense matmuls: TDM (tensor_load_to_lds) -> LDS -> V_WMMA_F32_16X16X4_F32,
// one wave per 16x16 tile, K processed in 64-wide TDM chunks.
// Edge segment-softmax: VALU + f32 global atomics.
// Attention (head_dim=2): LDS-tiled online softmax.
// ---------------------------------------------------------------------------

typedef __attribute__((ext_vector_type(2))) float v2f;
typedef __attribute__((ext_vector_type(8))) float v8f;
typedef __attribute__((ext_vector_type(4))) unsigned int u32x4;
typedef __attribute__((ext_vector_type(8))) int i32x8;
typedef __attribute__((ext_vector_type(4))) int i32x4;

#define N_NODES 4096
#define N_EDGES 131072
#define E_TOT   (N_EDGES + N_NODES)
#define F_IN    8
#define H1      4
#define C1      8
#define D1      32          // H1*C1
#define H2      4
#define C2      4
#define D_MODEL 16          // H2*C2
#define NHEAD   8
#define NLAYERS 2
#define DFF     2048
#define NGRAPHS 64
#define OUT_DIM 32
#define KCHUNK  64          // K elements per TDM tile

__device__ __forceinline__ float leakyf(float x, float s) {
    return x >= 0.f ? x : s * x;
}

__device__ __forceinline__ float atomicMaxF(float* addr, float val) {
    int* ai = (int*)addr;
    int old = __float_as_int(*addr);
    while (__int_as_float(old) < val) {
        int assumed = old;
        old = atomicCAS(ai, assumed, __float_as_int(val));
        if (old == assumed) break;
    }
    return __int_as_float(old);
}

__device__ __forceinline__ void edge_sd(const int* __restrict__ ei, int i, int& s, int& d) {
    if (i < N_EDGES) { s = ei[i]; d = ei[N_EDGES + i]; }
    else             { s = i - N_EDGES; d = s; }            // PyG-style self loops
}

// ---------------------------------------------------------------------------
// TDM: DMA a 2D tile [dim1 rows x dim0 contiguous elems] of f32 from global
// into LDS (packed row-major, row stride = dim0). D# per ISA 8.3/8.4:
//   group0: count=1 | lds_addr | global_addr(57b) | type=2
//   group1: data_size=4B, tensor_dim0/1 = tile dims (tile fully in-bounds),
//           tile_dim0/1, tensor_dim0_stride = strideElems. workgroup_mask=0.
// Tracked on TENSORcnt. 6-arg builtin form (clang-23 / therock headers).
// ---------------------------------------------------------------------------
__device__ __forceinline__ void tdm_load_2d(float* lds, const float* gptr,
                                            int dim0, int dim1, int strideElems)
{
    unsigned ldsOff = (unsigned)(uintptr_t)lds;          // low 32b = LDS offset
    unsigned long long ga = (unsigned long long)(uintptr_t)gptr;
    u32x4 g0;
    g0[0] = 1u;                                          // count=1, user mode
    g0[1] = ldsOff;                                      // lds_addr
    g0[2] = (unsigned)(ga & 0xffffffffu);                // global_addr[31:0]
    g0[3] = (unsigned)((ga >> 32) & 0x1ffffffu)          // global_addr[56:32]
          | (2u << 30);                                  // type=2 ("image")
    unsigned td0 = (unsigned)dim0, td1 = (unsigned)dim1;
    unsigned long long st = (unsigned long long)strideElems;
    i32x8 g1;
    g1[0] = (int)(2u << 16);                             // data_size = 4 bytes
    g1[1] = (int)((td0 & 0xffffu) << 16);                // tensor_dim0[15:0]
    g1[2] = (int)(((td0 >> 16) & 0xffffu)                // tensor_dim0[31:16]
          | ((td1 & 0xffffu) << 16));                    // tensor_dim1[15:0]
    g1[3] = (int)(((td1 >> 16) & 0xffffu)                // tensor_dim1[31:16]
          | (td0 << 16));                                // tile_dim0
    g1[4] = (int)(td1 & 0xffffu);                        // tile_dim1 (tile_dim2=0)
    g1[5] = (int)(st & 0xffffffffu);                     // tensor_dim0_stride lo
    g1[6] = (int)((st >> 32) & 0xffffu);                 // stride hi (dim1_stride=0)
    g1[7] = 0;
    i32x4 z4 = {0, 0, 0, 0};
    i32x8 z8 = {0, 0, 0, 0, 0, 0, 0, 0};
    __builtin_amdgcn_tensor_load_to_lds(g0, g1, z4, z4, z8, 0);
}

// ---------------------------------------------------------------------------
// fp32 WMMA GEMM: C[M,N] = A[M,K] @ B   (one wave = one 16x16 tile).
//   BT=false : B is [K,N] row-major.   BT=true : B is [N,K] row-major (W.T).
//   EPI: 0=store, 1=+bias, 2=+bias,relu, 3=+bias,+resid, 4=atomicAdd (split-K)
// K is streamed in KCHUNK tiles: TDM A-tile + B-tile -> s_wait_tensorcnt 0 ->
// ds_load fragments -> v_wmma_f32_16x16x4_f32.
// Fragment map (ISA 7.12.2): A 16x4: lanes0-15 K={0,1}, lanes16-31 K={2,3},
// row = lane&15; B mirrored (col = lane&15); C/D col = lane&15,
// rows = (lane>>4)*8 + r.
// Requires M%16==0, N%16==0, K%4==0, blockDim.x==32, no divergence.
// ---------------------------------------------------------------------------
template <bool BT, int EPI>
__global__ void k_wmma_gemm(const float* __restrict__ A, const float* __restrict__ B,
                            const float* __restrict__ bias, const float* __restrict__ resid,
                            float* __restrict__ C, int M, int Nn, int K, int kChunk)
{
    __shared__ float sA[16 * KCHUNK];
    __shared__ float sB[16 * KCHUNK];

    const int lane = threadIdx.x;
    const int hf   = lane >> 4;     // 0/1 : which K-pair / which row-half
    const int lid  = lane & 15;
    const int tn   = blockIdx.x << 4;
    const int tm   = blockIdx.y << 4;
    const int kBeg = blockIdx.z * kChunk;
    const int kEnd = kBeg + kChunk;

    v8f c = {0.f, 0.f, 0.f, 0.f, 0.f, 0.f, 0.f, 0.f};

    for (int k0 = kBeg; k0 < kEnd; k0 += KCHUNK) {
        const int chunk = (kEnd - k0) < KCHUNK ? (kEnd - k0) : KCHUNK;

        // A tile: 16 rows x chunk cols, row stride K
        tdm_load_2d(sA, A + (size_t)tm * K + k0, chunk, 16, K);
        // B tile
        if (BT) tdm_load_2d(sB, B + (size_t)tn * K + k0, chunk, 16, K);  // [n][k]
        else    tdm_load_2d(sB, B + (size_t)k0 * Nn + tn, 16, chunk, Nn); // [k][n]
        __builtin_amdgcn_s_wait_tensorcnt(0);

        for (int kk = 0; kk < chunk; kk += 4) {
            const int ka = kk + (hf << 1);
            v2f a, b;
            a.x = sA[lid * chunk + ka];
            a.y = sA[lid * chunk + ka + 1];
            if (BT) { b.x = sB[lid * chunk + ka];  b.y = sB[lid * chunk + ka + 1]; }
            else    { b.x = sB[ka * 16 + lid];     b.y = sB[(ka + 1) * 16 + lid]; }
            c = __builtin_amdgcn_wmma_f32_16x16x4_f32(false, a, false, b,
                                                      (short)0, c, false, false);
        }
        // ds_loads feeding the last WMMA are waited (DScnt) before its issue,
        // which precedes the next TDM overwrite in program order.
    }

    const int col = tn + lid;
    float bv = 0.f;
    if (EPI == 1 || EPI == 2 || EPI == 3) bv = bias[col];
#pragma unroll
    for (int r = 0; r < 8; ++r) {
        const int row = tm + (hf << 3) + r;
        if (EPI == 4) {
            atomicAdd(&C[row * Nn + col], c[r]);
        } else {
            float v = c[r] + bv;
            if (EPI == 2) v = v > 0.f ? v : 0.f;
            if (EPI == 3) v += resid[row * Nn + col];
            C[row * Nn + col] = v;
        }
    }
}

// ---------------------------------------------------------------------------
__global__ void k_fill(float* __restrict__ p, float v, int n)
{
    int i = blockIdx.x * blockDim.x + threadIdx.x;
    if (i < n) p[i] = v;
}

// per-node attention coefficients: as[n,h] = <h[n,h,:], att_src[h,:]>
__global__ void k_gat_coef(const float* __restrict__ h, const float* __restrict__ att_s,
                           const float* __restrict__ att_d, float* __restrict__ as_,
                           float* __restrict__ ad_, int heads, int ch)
{
    int nidx = blockIdx.x * blockDim.x + threadIdx.x;
    if (nidx >= N_NODES) return;
    const int dim = heads * ch;
    for (int hh = 0; hh < heads; ++hh) {
        float ss = 0.f, dd = 0.f;
        for (int cc = 0; cc < ch; ++cc) {
            float v = h[nidx * dim + hh * ch + cc];
            ss += v * att_s[hh * ch + cc];
            dd += v * att_d[hh * ch + cc];
        }
        as_[nidx * heads + hh] = ss;
        ad_[nidx * heads + hh] = dd;
    }
}

__global__ void k_edge_max(const int* __restrict__ ei, const float* __restrict__ as_,
                           const float* __restrict__ ad_, float* __restrict__ eb,
                           float* __restrict__ mx, int heads)
{
    int i = blockIdx.x * blockDim.x + threadIdx.x;
    if (i >= E_TOT) return;
    int s, d; edge_sd(ei, i, s, d);
    for (int hh = 0; hh < heads; ++hh) {
        float e = leakyf(as_[s * heads + hh] + ad_[d * heads + hh], 0.2f);
        eb[i * heads + hh] = e;
        atomicMaxF(&mx[d * heads + hh], e);
    }
}

__global__ void k_edge_exp(const int* __restrict__ ei, float* __restrict__ eb,
                           const float* __restrict__ mx, float* __restrict__ dn, int heads)
{
    int i = blockIdx.x * blockDim.x + threadIdx.x;
    if (i >= E_TOT) return;
    int s, d; edge_sd(ei, i, s, d); (void)s;
    for (int hh = 0; hh < heads; ++hh) {
        float p = __expf(eb[i * heads + hh] - mx[d * heads + hh]);
        eb[i * heads + hh] = p;
        atomicAdd(&dn[d * heads + hh], p);
    }
}

__global__ void k_edge_aggr(const int* __restrict__ ei, const float* __restrict__ eb,
                            const float* __restrict__ dn, const float* __restrict__ hsrc,
                            float* __restrict__ out, int heads, int ch)
{
    int i = blockIdx.x * blockDim.x + threadIdx.x;
    if (i >= E_TOT) return;
    int s, d; edge_sd(ei, i, s, d);
    const int dim = heads * ch;
    for (int hh = 0; hh < heads; ++hh) {
        float alpha = eb[i * heads + hh] / dn[d * heads + hh];
        for (int cc = 0; cc < ch; ++cc)
            atomicAdd(&out[d * dim + hh * ch + cc],
                      alpha * hsrc[s * dim + hh * ch + cc]);
    }
}

__global__ void k_bias_leaky(float* __restrict__ x, const float* __restrict__ bias,
                             int n, int dim, float slope)
{
    int i = blockIdx.x * blockDim.x + threadIdx.x;
    if (i < n * dim) x[i] = leakyf(x[i] + bias[i % dim], slope);
}

// ---------------------------------------------------------------------------
// LDS-tiled online-softmax attention. qkv: [S,48] = [q|k|v], head h owns dims
// {2h,2h+1}. Block = 256 thr = 32 queries x 8 heads; K/V streamed through LDS
// in 64-key tiles (8 KB).
// ---------------------------------------------------------------------------
#define ATT_TK 64
__global__ void k_attention(const float* __restrict__ qkv, float* __restrict__ ctx)
{
    __shared__ float skv[ATT_TK][32];       // [key][ k(16) | v(16) ]
    const int tid  = threadIdx.x;
    const int qloc = tid >> 3;
    const int hh   = tid & 7;
    const int q    = blockIdx.x * 32 + qloc;
    const float scale = 0.70710678118654752f;   // 1/sqrt(head_dim=2)
    const float q0 = qkv[q * 48 + hh * 2];
    const float q1 = qkv[q * 48 + hh * 2 + 1];
    float m = -3.0e38f, l = 0.f, a0 = 0.f, a1 = 0.f;

    for (int kt = 0; kt < N_NODES; kt += ATT_TK) {
        for (int i = tid; i < ATT_TK * 32; i += 256) {
            int key = i >> 5, cc = i & 31;
            skv[key][cc] = qkv[(kt + key) * 48 + 16 + cc];
        }
        __syncthreads();
#pragma unroll 4
        for (int kk = 0; kk < ATT_TK; ++kk) {
            float s    = (q0 * skv[kk][hh * 2] + q1 * skv[kk][hh * 2 + 1]) * scale;
            float mn   = fmaxf(m, s);
            float corr = __expf(m - mn);
            float p    = __expf(s - mn);
            l  = l  * corr + p;
            a0 = a0 * corr + p * skv[kk][16 + hh * 2];
            a1 = a1 * corr + p * skv[kk][17 + hh * 2];
            m  = mn;
        }
        __syncthreads();
    }
    ctx[q * 16 + hh * 2]     = a0 / l;
    ctx[q * 16 + hh * 2 + 1] = a1 / l;
}

__global__ void k_layernorm(const float* __restrict__ in, const float* __restrict__ sc,
                            const float* __restrict__ bi, float* __restrict__ out)
{
    int r = blockIdx.x * blockDim.x + threadIdx.x;
    if (r >= N_NODES) return;
    float mu = 0.f;
    for (int c = 0; c < D_MODEL; ++c) mu += in[r * D_MODEL + c];
    mu *= (1.f / D_MODEL);
    float var = 0.f;
    for (int c = 0; c < D_MODEL; ++c) {
        float dv = in[r * D_MODEL + c] - mu;
        var += dv * dv;
    }
    var *= (1.f / D_MODEL);
    float inv = rsqrtf(var + 1e-5f);
    for (int c = 0; c < D_MODEL; ++c)
        out[r * D_MODEL + c] = (in[r * D_MODEL + c] - mu) * inv * sc[c] + bi[c];
}

// seed C = resid + bias (accumulation target for split-K FF2)
__global__ void k_seed(float* __restrict__ C, const float* __restrict__ resid,
                       const float* __restrict__ bias, int n, int dim)
{
    int i = blockIdx.x * blockDim.x + threadIdx.x;
    if (i < n * dim) C[i] = resid[i] + bias[i % dim];
}

__global__ void k_pool(const float* __restrict__ h, const int* __restrict__ batch,
                       float* __restrict__ pooled, float* __restrict__ cnt)
{
    int i = blockIdx.x * blockDim.x + threadIdx.x;
    if (i >= N_NODES) return;
    int g = batch[i];
    atomicAdd(&cnt[g], 1.f);
    for (int c = 0; c < D_MODEL; ++c)
        atomicAdd(&pooled[g * D_MODEL + c], h[i * D_MODEL + c]);
}

__global__ void k_pooldiv(float* __restrict__ pooled, const float* __restrict__ cnt)
{
    int i = blockIdx.x * blockDim.x + threadIdx.x;
    if (i < NGRAPHS * D_MODEL) pooled[i] /= fmaxf(cnt[i / D_MODEL], 1.f);
}

// ---------------------------------------------------------------------------
static inline dim3 g1d(int n, int b) { return dim3((n + b - 1) / b); }

extern "C" void kernel_launch(void* const* d_in, const int* in_sizes, int n_in,
                              void* d_out, int out_size, void* d_ws, size_t ws_size,
                              hipStream_t stream)
{
    (void)in_sizes; (void)n_in; (void)out_size; (void)ws_size;
    const float* x    = (const float*)d_in[0];
    const int*   ei   = (const int*)  d_in[1];
    const int*   bat  = (const int*)  d_in[2];
    const float* g1w  = (const float*)d_in[3];
    const float* g1as = (const float*)d_in[4];
    const float* g1ad = (const float*)d_in[5];
    const float* g1b  = (const float*)d_in[6];
    const float* g2w  = (const float*)d_in[7];
    const float* g2as = (const float*)d_in[8];
    const float* g2ad = (const float*)d_in[9];
    const float* g2b  = (const float*)d_in[10];
    const float* inw  = (const float*)d_in[11];
    const float* inb  = (const float*)d_in[12];
    const float* outw = (const float*)d_in[13];
    const float* outb = (const float*)d_in[14];
    const float* f1w  = (const float*)d_in[15];
    const float* f1b  = (const float*)d_in[16];
    const float* f2w  = (const float*)d_in[17];
    const float* f2b  = (const float*)d_in[18];
    const float* ln1s = (const float*)d_in[19];
    const float* ln1b = (const float*)d_in[20];
    const float* ln2s = (const float*)d_in[21];
    const float* ln2b = (const float*)d_in[22];
    const float* fcw  = (const float*)d_in[23];
    const float* fcb  = (const float*)d_in[24];
    float* out = (float*)d_out;

    // ---- workspace carve (~41 MB of f32) ----
    float* w = (float*)d_ws;
    float* h1   = w; w += N_NODES * D1;
    float* as1  = w; w += N_NODES * H1;
    float* ad1  = w; w += N_NODES * H1;
    float* mx1  = w; w += N_NODES * H1;
    float* dn1  = w; w += N_NODES * H1;
    float* eb1  = w; w += E_TOT * H1;
    float* gout = w; w += N_NODES * D1;      // GAT1 output
    float* h2   = w; w += N_NODES * D_MODEL; // GAT2 transformed feats
    float* as2  = w; w += N_NODES * H2;
    float* ad2  = w; w += N_NODES * H2;
    float* mx2  = w; w += N_NODES * H2;
    float* dn2  = w; w += N_NODES * H2;
    float* eb2  = w; w += E_TOT * H2;
    float* hbuf = w; w += N_NODES * D_MODEL; // transformer state
    float* qkv  = w; w += N_NODES * 3 * D_MODEL;
    float* ctx  = w; w += N_NODES * D_MODEL;
    float* tmp  = w; w += N_NODES * D_MODEL;
    float* mid  = w; w += (size_t)N_NODES * DFF; // FF hidden, 32 MB
    float* pool = w; w += NGRAPHS * D_MODEL;
    float* cnt  = w; w += NGRAPHS;

    // ======================= GAT layer 1 =======================
    // h1 = x @ gat1_w   [4096,8] x [8,32]  (B row-major [K,N])
    k_wmma_gemm<false, 0><<<dim3(D1 / 16, N_NODES / 16, 1), 32, 0, stream>>>(
        x, g1w, nullptr, nullptr, h1, N_NODES, D1, F_IN, F_IN);
    k_gat_coef<<<g1d(N_NODES, 256), 256, 0, stream>>>(h1, g1as, g1ad, as1, ad1, H1, C1);
    k_fill<<<g1d(N_NODES * H1, 256), 256, 0, stream>>>(mx1, -3.0e38f, N_NODES * H1);
    k_fill<<<g1d(N_NODES * H1, 256), 256, 0, stream>>>(dn1, 0.f, N_NODES * H1);
    k_fill<<<g1d(N_NODES * D1, 256), 256, 0, stream>>>(gout, 0.f, N_NODES * D1);
    k_edge_max <<<g1d(E_TOT, 256), 256, 0, stream>>>(ei, as1, ad1, eb1, mx1, H1);
    k_edge_exp <<<g1d(E_TOT, 256), 256, 0, stream>>>(ei, eb1, mx1, dn1, H1);
    k_edge_aggr<<<g1d(E_TOT, 256), 256, 0, stream>>>(ei, eb1, dn1, h1, gout, H1, C1);
    k_bias_leaky<<<g1d(N_NODES * D1, 256), 256, 0, stream>>>(gout, g1b, N_NODES, D1, 0.01f);

    // ======================= GAT layer 2 =======================
    k_wmma_gemm<false, 0><<<dim3(D_MODEL / 16, N_NODES / 16, 1), 32, 0, stream>>>(
        gout, g2w, nullptr, nullptr, h2, N_NODES, D_MODEL, D1, D1);
    k_gat_coef<<<g1d(N_NODES, 256), 256, 0, stream>>>(h2, g2as, g2ad, as2, ad2, H2, C2);
    k_fill<<<g1d(N_NODES * H2, 256), 256, 0, stream>>>(mx2, -3.0e38f, N_NODES * H2);
    k_fill<<<g1d(N_NODES * H2, 256), 256, 0, stream>>>(dn2, 0.f, N_NODES * H2);
    k_fill<<<g1d(N_NODES * D_MODEL, 256), 256, 0, stream>>>(hbuf, 0.f, N_NODES * D_MODEL);
    k_edge_max <<<g1d(E_TOT, 256), 256, 0, stream>>>(ei, as2, ad2, eb2, mx2, H2);
    k_edge_exp <<<g1d(E_TOT, 256), 256, 0, stream>>>(ei, eb2, mx2, dn2, H2);
    k_edge_aggr<<<g1d(E_TOT, 256), 256, 0, stream>>>(ei, eb2, dn2, h2, hbuf, H2, C2);
    k_bias_leaky<<<g1d(N_NODES * D_MODEL, 256), 256, 0, stream>>>(hbuf, g2b, N_NODES, D_MODEL, 0.01f);

    // =================== Transformer layers ====================
    for (int l = 0; l < NLAYERS; ++l) {
        const float* Win  = inw  + (size_t)l * 3 * D_MODEL * D_MODEL; // [48,16]
        const float* Wout = outw + (size_t)l * D_MODEL * D_MODEL;     // [16,16]
        const float* W1   = f1w  + (size_t)l * DFF * D_MODEL;         // [2048,16]
        const float* W2   = f2w  + (size_t)l * D_MODEL * DFF;         // [16,2048]

        // qkv = h @ in_w.T + in_b   [4096,16] x [16,48]
        k_wmma_gemm<true, 1><<<dim3(3 * D_MODEL / 16, N_NODES / 16, 1), 32, 0, stream>>>(
            hbuf, Win, inb + l * 3 * D_MODEL, nullptr, qkv, N_NODES, 3 * D_MODEL, D_MODEL, D_MODEL);

        k_attention<<<dim3(N_NODES / 32), 256, 0, stream>>>(qkv, ctx);

        // tmp = ctx @ out_w.T + out_b + h (residual)
        k_wmma_gemm<true, 3><<<dim3(1, N_NODES / 16, 1), 32, 0, stream>>>(
            ctx, Wout, outb + l * D_MODEL, hbuf, tmp, N_NODES, D_MODEL, D_MODEL, D_MODEL);
        k_layernorm<<<g1d(N_NODES, 256), 256, 0, stream>>>(tmp, ln1s + l * D_MODEL,
                                                           ln1b + l * D_MODEL, hbuf);

        // mid = relu(h @ ff1_w.T + ff1_b)   [4096,16] x [16,2048]
        k_wmma_gemm<true, 2><<<dim3(DFF / 16, N_NODES / 16, 1), 32, 0, stream>>>(
            hbuf, W1, f1b + l * DFF, nullptr, mid, N_NODES, DFF, D_MODEL, D_MODEL);

        // tmp = h + ff2_b ; tmp += mid @ ff2_w.T (split-K x32, atomic f32 adds)
        k_seed<<<g1d(N_NODES * D_MODEL, 256), 256, 0, stream>>>(
            tmp, hbuf, f2b + l * D_MODEL, N_NODES, D_MODEL);
        k_wmma_gemm<true, 4><<<dim3(1, N_NODES / 16, DFF / KCHUNK), 32, 0, stream>>>(
            mid, W2, nullptr, nullptr, tmp, N_NODES, D_MODEL, DFF, KCHUNK);
        k_layernorm<<<g1d(N_NODES, 256), 256, 0, stream>>>(tmp, ln2s + l * D_MODEL,
                                                           ln2b + l * D_MODEL, hbuf);
    }

    // ================= Pool + final linear =====================
    k_fill<<<g1d(NGRAPHS * D_MODEL, 64), 64, 0, stream>>>(pool, 0.f, NGRAPHS * D_MODEL);
    k_fill<<<g1d(NGRAPHS, 64), 64, 0, stream>>>(cnt, 0.f, NGRAPHS);
    k_pool<<<g1d(N_NODES, 256), 256, 0, stream>>>(hbuf, bat, pool, cnt);
    k_pooldiv<<<g1d(NGRAPHS * D_MODEL, 64), 64, 0, stream>>>(pool, cnt);

    // out = pooled @ fc_w.T + fc_b   [64,16] x [16,32]
    k_wmma_gemm<true, 1><<<dim3(OUT_DIM / 16, NGRAPHS / 16, 1), 32, 0, stream>>>(
        pool, fcw, fcb, nullptr, out, NGRAPHS, OUT_DIM, D_MODEL, D_MODEL);
}